// KGAT_91336774517081
// MI455X (gfx1250) — compile-verified
//
#include <hip/hip_runtime.h>
#include <hip/hip_bf16.h>
#include <math.h>

typedef __attribute__((ext_vector_type(2))) float v2f;
typedef __attribute__((ext_vector_type(8))) float v8f;

#define DD 64
#define HH 4
#define HD 256
#define NEG_SLOPE 0.2f
#define EPSA 1e-16f

// -------- monotone float<->uint key for atomic max on arbitrary-sign floats ----
__device__ __forceinline__ unsigned f2key(float f) {
    unsigned b = __float_as_uint(f);
    return (b & 0x80000000u) ? ~b : (b | 0x80000000u);
}
__device__ __forceinline__ float key2f(unsigned k) {
    unsigned b = (k & 0x80000000u) ? (k ^ 0x80000000u) : ~k;
    return __uint_as_float(b);
}

// ================= K1: xp = entity_table[N,64] @ lin_w[64,256]  (fp32 WMMA) ====
// one wave per 16x16 output tile; 16 chained V_WMMA_F32_16X16X4_F32 over K=64.
__global__ void __launch_bounds__(32)
k_gemm_xp(const float* __restrict__ ent, const float* __restrict__ lw,
          float* __restrict__ xp, int N) {
    const int mt   = blockIdx.x;          // row tile (16 rows)
    const int col0 = blockIdx.y * 16;     // col tile (16 cols of 256)
    const int lane = threadIdx.x;         // 0..31
    const int m    = lane & 15;
    const int kh   = (lane >> 4) * 2;     // 0 (lanes 0-15) or 2 (lanes 16-31)
    int row = mt * 16 + m;
    if (row >= N) row = N - 1;            // clamp loads; stores guarded below

    v8f c = {0.f,0.f,0.f,0.f,0.f,0.f,0.f,0.f};
    const float* arow = ent + (size_t)row * DD;
#pragma unroll
    for (int k0 = 0; k0 < DD; k0 += 4) {
        // A 16x4 tile: lane m holds A[m][kh], A[m][kh+1]
        v2f a = *(const v2f*)(arow + k0 + kh);
        // B 4x16 tile: lane holds B[kh][n], B[kh+1][n], n = lane&15
        v2f b;
        b.x = lw[(size_t)(k0 + kh    ) * HD + col0 + m];
        b.y = lw[(size_t)(k0 + kh + 1) * HD + col0 + m];
        c = __builtin_amdgcn_wmma_f32_16x16x4_f32(false, a, false, b,
                                                  (short)0, c, false, false);
    }
    // C/D layout: VGPR j -> row (j or j+8), col = lane&15
    const int rbase = mt * 16 + ((lane < 16) ? 0 : 8);
#pragma unroll
    for (int j = 0; j < 8; ++j) {
        int r = rbase + j;
        if (r < N) xp[(size_t)r * HD + col0 + m] = c[j];
    }
}

// ================= K2: a_src[n,h], a_dst[n,h] = <xp[n,h,:], att_{src,dst}[h,:]> ==
__global__ void k_attn(const float* __restrict__ xp,
                       const float* __restrict__ att_src,
                       const float* __restrict__ att_dst,
                       float* __restrict__ a_src, float* __restrict__ a_dst, int N) {
    int idx = blockIdx.x * blockDim.x + threadIdx.x;   // n*H + h
    if (idx >= N * HH) return;
    int h = idx & (HH - 1);
    const float* xr = xp + (size_t)(idx >> 2) * HD + h * DD;
    const float* as = att_src + h * DD;
    const float* ad = att_dst + h * DD;
    float s = 0.f, d = 0.f;
#pragma unroll 8
    for (int cidx = 0; cidx < DD; ++cidx) {
        float v = xr[cidx];
        s = fmaf(v, as[cidx], s);
        d = fmaf(v, ad[cidx], d);
    }
    a_src[idx] = s;
    a_dst[idx] = d;
}

// ================= K3: per-edge leaky-relu logits + scatter max over dst =======
__global__ void k_edge_logits(const int* __restrict__ ei, int E, int N,
                              const float* __restrict__ a_src,
                              const float* __restrict__ a_dst,
                              float* __restrict__ logits,
                              unsigned* __restrict__ maxkey) {
    int e = blockIdx.x * blockDim.x + threadIdx.x;
    int Et = E + N;
    if (e >= Et) return;
    int s, d;
    if (e < E) { s = ei[e]; d = ei[E + e]; } else { s = d = e - E; }
#pragma unroll
    for (int h = 0; h < HH; ++h) {
        float l = a_src[s * HH + h] + a_dst[d * HH + h];
        l = (l > 0.f) ? l : NEG_SLOPE * l;
        logits[(size_t)e * HH + h] = l;
        atomicMax(&maxkey[d * HH + h], f2key(l));
    }
}

// ================= K4: e = exp(l - mx[dst]); denom[dst] += e (in-place) ========
__global__ void k_edge_exp(const int* __restrict__ ei, int E, int N,
                           float* __restrict__ ebuf,           // in: logits, out: exp
                           const unsigned* __restrict__ maxkey,
                           float* __restrict__ denom) {
    int e = blockIdx.x * blockDim.x + threadIdx.x;
    int Et = E + N;
    if (e >= Et) return;
    int d = (e < E) ? ei[E + e] : (e - E);
#pragma unroll
    for (int h = 0; h < HH; ++h) {
        float mx = key2f(maxkey[d * HH + h]);   // always finite (self-loops)
        float ex = expf(ebuf[(size_t)e * HH + h] - mx);
        ebuf[(size_t)e * HH + h] = ex;
        atomicAdd(&denom[d * HH + h], ex);
    }
}

// ================= K5: out[dst,h,:] += alpha * xp[src,h,:]   (wave per edge) ===
__global__ void __launch_bounds__(256)
k_edge_aggregate(const int* __restrict__ ei, int E, int N,
                 const float* __restrict__ ebuf,
                 const float* __restrict__ denom,
                 const float* __restrict__ xp,
                 float* __restrict__ outagg) {
    int wid  = blockIdx.x * (blockDim.x >> 5) + (threadIdx.x >> 5);  // edge id
    int lane = threadIdx.x & 31;
    int Et = E + N;
    if (wid >= Et) return;
    int s, d;
    if (wid < E) { s = ei[wid]; d = ei[E + wid]; } else { s = d = wid - E; }
    int h = lane >> 3;                       // 8 lanes per head (8 channels each)
    float alpha = ebuf[(size_t)wid * HH + h] / (denom[d * HH + h] + EPSA);
    const float4* xs = (const float4*)(xp + (size_t)s * HD + lane * 8);
    float4 v0 = xs[0], v1 = xs[1];
    float* po = outagg + (size_t)d * HD + lane * 8;
    atomicAdd(po + 0, v0.x * alpha);
    atomicAdd(po + 1, v0.y * alpha);
    atomicAdd(po + 2, v0.z * alpha);
    atomicAdd(po + 3, v0.w * alpha);
    atomicAdd(po + 4, v1.x * alpha);
    atomicAdd(po + 5, v1.y * alpha);
    atomicAdd(po + 6, v1.z * alpha);
    atomicAdd(po + 7, v1.w * alpha);
}

// ================= K6: x[n,c] = mean_h out[n,h,c] + bias[c] ====================
__global__ void k_finalize(const float* __restrict__ outagg,
                           const float* __restrict__ bias,
                           float* __restrict__ xf, int N) {
    int idx = blockIdx.x * blockDim.x + threadIdx.x;   // n*64 + c
    if (idx >= N * DD) return;
    int n = idx >> 6, cc = idx & 63;
    const float* r = outagg + (size_t)n * HD;
    xf[idx] = 0.25f * (r[cc] + r[DD + cc] + r[2 * DD + cc] + r[3 * DD + cc]) + bias[cc];
}

// ================= K7: sigmoid( [u|v] @ pred_w + b ) ==========================
__global__ void k_predict(const float* __restrict__ xf,
                          const int* __restrict__ ui, const int* __restrict__ vi,
                          const float* __restrict__ pw, const float* __restrict__ pb,
                          float* __restrict__ out, int B) {
    int b = blockIdx.x * blockDim.x + threadIdx.x;
    if (b >= B) return;
    const float* u = xf + (size_t)ui[b] * DD;
    const float* v = xf + (size_t)vi[b] * DD;
    float s = pb[0];
#pragma unroll 8
    for (int cidx = 0; cidx < DD; ++cidx) {
        s = fmaf(u[cidx], pw[cidx], s);
        s = fmaf(v[cidx], pw[DD + cidx], s);
    }
    out[b] = 1.f / (1.f + expf(-s));
}

// =============================================================================
extern "C" void kernel_launch(void* const* d_in, const int* in_sizes, int n_in,
                              void* d_out, int out_size, void* d_ws, size_t ws_size,
                              hipStream_t stream) {
    (void)n_in; (void)ws_size;
    const float* ent      = (const float*)d_in[0];   // [N,64]
    const float* lw       = (const float*)d_in[2];   // [64,256]
    const float* att_src  = (const float*)d_in[3];   // [4,64]
    const float* att_dst  = (const float*)d_in[4];   // [4,64]
    const float* gbias    = (const float*)d_in[5];   // [64]
    const float* pw       = (const float*)d_in[6];   // [128,1]
    const float* pb       = (const float*)d_in[7];   // [1]
    const int*   ei       = (const int*)d_in[8];     // [2,E] flat
    const int*   ui       = (const int*)d_in[10];    // [B]
    const int*   vi       = (const int*)d_in[11];    // [B]
    float* out = (float*)d_out;

    const int N  = in_sizes[0] / DD;
    const int E  = in_sizes[8] / 2;
    const int B  = in_sizes[10];
    const int Et = E + N;

    // ---- workspace carve-up (256B aligned) ----
    char* ws = (char*)d_ws;
    size_t off = 0;
    auto carve = [&](size_t bytes) {
        void* p = ws + off;
        off += (bytes + 255) & ~(size_t)255;
        return p;
    };
    float*    xp     = (float*)   carve((size_t)N * HD * 4);  // 51.2 MB
    float*    outagg = (float*)   carve((size_t)N * HD * 4);  // 51.2 MB
    float*    a_src  = (float*)   carve((size_t)N * HH * 4);
    float*    a_dst  = (float*)   carve((size_t)N * HH * 4);
    unsigned* maxkey = (unsigned*)carve((size_t)N * HH * 4);
    float*    denom  = (float*)   carve((size_t)N * HH * 4);
    float*    ebuf   = (float*)   carve((size_t)Et * HH * 4); // 13.6 MB
    float*    xf     = (float*)   carve((size_t)N * DD * 4);  // 12.8 MB

    // ---- zero the accumulators (graph-capture-safe async memsets) ----
    hipMemsetAsync(outagg, 0, (size_t)N * HD * 4, stream);
    hipMemsetAsync(maxkey, 0, (size_t)N * HH * 4, stream);  // 0 < key of any real
    hipMemsetAsync(denom,  0, (size_t)N * HH * 4, stream);

    // K1: GEMM via fp32 WMMA
    {
        dim3 grid((N + 15) / 16, HD / 16);
        k_gemm_xp<<<grid, 32, 0, stream>>>(ent, lw, xp, N);
    }
    // K2: attention dots
    {
        int t = N * HH;
        k_attn<<<(t + 255) / 256, 256, 0, stream>>>(xp, att_src, att_dst, a_src, a_dst, N);
    }
    // K3: logits + scatter-max
    k_edge_logits<<<(Et + 255) / 256, 256, 0, stream>>>(ei, E, N, a_src, a_dst, ebuf, maxkey);
    // K4: exp + scatter-sum denom
    k_edge_exp<<<(Et + 255) / 256, 256, 0, stream>>>(ei, E, N, ebuf, maxkey, denom);
    // K5: weighted message aggregation (one wave32 per edge)
    {
        int wavesPerBlock = 256 / 32;
        int blocks = (Et + wavesPerBlock - 1) / wavesPerBlock;
        k_edge_aggregate<<<blocks, 256, 0, stream>>>(ei, E, N, ebuf, denom, xp, outagg);
    }
    // K6: head mean + bias
    {
        int t = N * DD;
        k_finalize<<<(t + 255) / 256, 256, 0, stream>>>(outagg, gbias, xf, N);
    }
    // K7: link prediction
    k_predict<<<(B + 255) / 256, 256, 0, stream>>>(xf, ui, vi, pw, pb, out, B);
}